// BattleEmbeddingModule_42855183680030
// MI455X (gfx1250) — compile-verified
//
#include <hip/hip_runtime.h>

typedef __attribute__((ext_vector_type(2))) float v2f;
typedef __attribute__((ext_vector_type(8))) float v8f;

#define BATCH      32768
#define IN_STRIDE  10451
#define OUT_STRIDE 542

// LDS layout per weight: [(Kpad/2)][Npad][2] floats, zero-padded.
#define OFF_PLAYER 0       // K=9   Kpad=12  Npad=16 -> 192
#define OFF_STATUS 192     // K=86  Kpad=88  Npad=16 -> 1408
#define OFF_PINFO  1600    // K=8   Kpad=8   Npad=16 -> 128
#define OFF_CARD   1728    // K=740 Kpad=740 Npad=32 -> 23680
#define OFF_POTION 25408   // K=43  Kpad=44  Npad=16 -> 704
#define OFF_RELICS 26112   // K=180 Kpad=180 Npad=16 -> 2880
#define OFF_MONST  28992   // K=101 Kpad=104 Npad=16 -> 1664
#define LDS_FLOATS 30656   // 122,624 bytes (< 320 KB WGP LDS)

__device__ __forceinline__ void tile_gemm(
    const float* __restrict__ battle, float* __restrict__ out,
    int row0, int in_off, int K, const float* lwt, int Npad, int N, int out_off)
{
    const int lane = threadIdx.x & 31;
    const int mrow = lane & 15;      // A: lane -> matrix row M
    const int hi   = lane >> 4;      // A: VGPR half selects K pair 0/1 vs 2/3
    const float* arow = battle + (size_t)(row0 + mrow) * IN_STRIDE + in_off;
    const int K4 = K & ~3;

    for (int nt = 0; nt < Npad; nt += 16) {
        const int n = (lane & 15) + nt;
        v8f acc = {};
        int k = 0;
        for (; k < K4; k += 4) {
            const int kq = k + hi * 2;
            v2f a;
            a[0] = arow[kq];
            a[1] = arow[kq + 1];
            // B fragment: one 64-bit LDS load of W^T[kq, n], W^T[kq+1, n]
            v2f b = *(const v2f*)&lwt[(size_t)(kq >> 1) * (Npad * 2) + n * 2];
            acc = __builtin_amdgcn_wmma_f32_16x16x4_f32(
                false, a, false, b, (short)0, acc, false, false);
            if ((k & 124) == 0)
                __builtin_prefetch(arow + k + 256, 0, 0);
        }
        if (K & 3) { // tail: zero-pad A per lane (no EXEC change for the WMMA)
            const int kq = k + hi * 2;
            v2f a;
            a[0] = (kq     < K) ? arow[kq]     : 0.0f;
            a[1] = (kq + 1 < K) ? arow[kq + 1] : 0.0f;
            v2f b = *(const v2f*)&lwt[(size_t)(kq >> 1) * (Npad * 2) + n * 2];
            acc = __builtin_amdgcn_wmma_f32_16x16x4_f32(
                false, a, false, b, (short)0, acc, false, false);
        }
        if (n < N) { // C/D layout: VGPR r -> M = r + hi*8, N = lane&15
            float* orow = out + (size_t)(row0 + hi * 8) * OUT_STRIDE + out_off + n;
#pragma unroll
            for (int r = 0; r < 8; ++r)
                orow[(size_t)r * OUT_STRIDE] = acc[r];
        }
    }
}

__global__ void __launch_bounds__(256) battle_embed_kernel(
    const float* __restrict__ battle, float* __restrict__ out,
    const float* __restrict__ Wp, const float* __restrict__ Ws,
    const float* __restrict__ Wi, const float* __restrict__ Wc,
    const float* __restrict__ Wo, const float* __restrict__ Wr,
    const float* __restrict__ Wm)
{
    extern __shared__ float lds[];

    struct Meta { const float* W; int K; int N; int Kpad; int Npad; int off; };
    const Meta metas[7] = {
        {Wp,   9,  4,  12, 16, OFF_PLAYER},
        {Ws,  86, 16,  88, 16, OFF_STATUS},
        {Wi,   8,  2,   8, 16, OFF_PINFO },
        {Wc, 740, 32, 740, 32, OFF_CARD  },
        {Wo,  43,  8,  44, 16, OFF_POTION},
        {Wr, 180, 16, 180, 16, OFF_RELICS},
        {Wm, 101, 16, 104, 16, OFF_MONST },
    };

    // Cooperative transpose of all weights into LDS (zero-padded).
    for (int u = 0; u < 7; ++u) {
        const Meta mt = metas[u];
        const int total = mt.Kpad * mt.Npad;
        for (int idx = threadIdx.x; idx < total; idx += 256) {
            const int pairIdx = idx / (mt.Npad * 2);
            const int rem = idx - pairIdx * (mt.Npad * 2);
            const int n = rem >> 1;
            const int k = pairIdx * 2 + (rem & 1);
            float v = 0.0f;
            if (k < mt.K && n < mt.N) v = mt.W[n * mt.K + k];
            lds[mt.off + idx] = v;
        }
    }
    __syncthreads();

    const int wave = threadIdx.x >> 5;
    const int row0 = blockIdx.x * 128 + wave * 16;

    tile_gemm(battle, out, row0,    0,   9, lds + OFF_PLAYER, 16,  4,   0);
    tile_gemm(battle, out, row0,    9,  86, lds + OFF_STATUS, 16, 16,   4);
    tile_gemm(battle, out, row0,   95,   8, lds + OFF_PINFO,  16,  2,  20);
    for (int s = 0; s < 13; ++s)
        tile_gemm(battle, out, row0, 103 + s * 740, 740, lds + OFF_CARD, 32, 32,
                  22 + s * 32);
    tile_gemm(battle, out, row0, 9723,  43, lds + OFF_POTION, 16,  8, 438);
    tile_gemm(battle, out, row0, 9766, 180, lds + OFF_RELICS, 16, 16, 446);
    for (int s = 0; s < 5; ++s)
        tile_gemm(battle, out, row0, 9946 + s * 101, 101, lds + OFF_MONST, 16, 16,
                  462 + s * 16);
}

extern "C" void kernel_launch(void* const* d_in, const int* in_sizes, int n_in,
                              void* d_out, int out_size, void* d_ws, size_t ws_size,
                              hipStream_t stream) {
    const float* battle = (const float*)d_in[0];
    const float* Wp = (const float*)d_in[1];
    const float* Ws = (const float*)d_in[2];
    const float* Wi = (const float*)d_in[3];
    const float* Wc = (const float*)d_in[4];
    const float* Wo = (const float*)d_in[5];
    const float* Wr = (const float*)d_in[6];
    const float* Wm = (const float*)d_in[7];
    float* out = (float*)d_out;

    dim3 grid(BATCH / 128);   // 256 blocks, 8 waves each, 16 rows per wave
    dim3 block(256);
    battle_embed_kernel<<<grid, block, LDS_FLOATS * sizeof(float), stream>>>(
        battle, out, Wp, Ws, Wi, Wc, Wo, Wr, Wm);
}